// CPN_WCP_44908178047551
// MI455X (gfx1250) — compile-verified
//
#include <hip/hip_runtime.h>
#include <hip/hip_bf16.h>
#include <math.h>

// Problem constants (match reference setup_inputs)
#define AUGN   4
#define BATCH  128
#define DFEAT  512
#define NTOT   (AUGN * BATCH)           // 512
#define MPROB  (AUGN * BATCH * AUGN)    // 2048
#define CPAD   129                      // LDS row stride for conflict-free col access

typedef float v2f __attribute__((ext_vector_type(2)));
typedef float v8f __attribute__((ext_vector_type(8)));

// ---------------- wave32 butterfly reductions ----------------
__device__ __forceinline__ float wave_max32(float v) {
  #pragma unroll
  for (int m = 16; m >= 1; m >>= 1) v = fmaxf(v, __shfl_xor(v, m, 32));
  return v;
}
__device__ __forceinline__ float wave_min32(float v) {
  #pragma unroll
  for (int m = 16; m >= 1; m >>= 1) v = fminf(v, __shfl_xor(v, m, 32));
  return v;
}
__device__ __forceinline__ float wave_sum32(float v) {
  #pragma unroll
  for (int m = 16; m >= 1; m >>= 1) v += __shfl_xor(v, m, 32);
  return v;
}

// ---------------- out zeroing (harness poisons d_out) ----------------
__global__ void zero_out_kernel(float* out) { out[0] = 0.0f; }

// ---------------- Gram: dst[n x n] = src[n x k] * src^T via fp32 WMMA ----------------
// One wave per 16x16 output tile. A is 16x4 (MxK), B is 4x16 (KxN) per WMMA step.
// A frag layout (32-bit A 16x4): lane = 16*half + m ; vgpr0 -> K = 2*half, vgpr1 -> K = 2*half+1
// B frag layout mirrors (B[k][n] = src[n][k]), so loading is identical with the N-tile row.
__global__ __launch_bounds__(32) void gram_wmma_kernel(const float* __restrict__ src,
                                                       float* __restrict__ dst,
                                                       int n, int k) {
  const int tn   = blockIdx.x;       // N tile
  const int tm   = blockIdx.y;       // M tile
  const int lane = threadIdx.x;      // 0..31
  const int half = lane >> 4;        // 0/1
  const int lm   = lane & 15;        // row-in-tile

  const float2* a2 = reinterpret_cast<const float2*>(src + (size_t)(tm * 16 + lm) * k);
  const float2* b2 = reinterpret_cast<const float2*>(src + (size_t)(tn * 16 + lm) * k);

  v8f c = {};
  for (int kk = 0; kk < k; kk += 32) {
    #pragma unroll
    for (int uu = 0; uu < 8; ++uu) {
      const int kb = (kk + 4 * uu) >> 1;        // float2 index of K-base
      float2 af = a2[kb + half];                // (K=2*half, K=2*half+1)
      float2 bf = b2[kb + half];
      v2f a; a[0] = af.x; a[1] = af.y;
      v2f b; b[0] = bf.x; b[1] = bf.y;
      c = __builtin_amdgcn_wmma_f32_16x16x4_f32(false, a, false, b, (short)0, c,
                                                false, false);
    }
  }
  // C/D layout: vgpr g, lanes 0-15 -> M=g ; lanes 16-31 -> M=g+8 ; N = lane&15
  #pragma unroll
  for (int g = 0; g < 8; ++g) {
    dst[(size_t)(tm * 16 + g + 8 * half) * n + (tn * 16 + lm)] = c[g];
  }
}

// ---------------- fn = normalize(mean over aug of features) : [BATCH x DFEAT] ----------------
__global__ __launch_bounds__(256) void fn_kernel(const float* __restrict__ feat,
                                                 float* __restrict__ fn) {
  __shared__ float red[8];
  const int b = blockIdx.x;               // 0..127
  const int tid = threadIdx.x;            // 0..255
  const int lane = tid & 31, wave = tid >> 5;

  float f0 = 0.f, f1 = 0.f;
  const int k0 = tid, k1 = tid + 256;
  #pragma unroll
  for (int a = 0; a < AUGN; ++a) {
    const float* row = feat + (size_t)(a * BATCH + b) * DFEAT;
    f0 += row[k0];
    f1 += row[k1];
  }
  f0 *= (1.0f / AUGN);
  f1 *= (1.0f / AUGN);

  float ss = f0 * f0 + f1 * f1;
  ss = wave_sum32(ss);
  if (lane == 0) red[wave] = ss;
  __syncthreads();
  float tot = 0.f;
  #pragma unroll
  for (int w = 0; w < 8; ++w) tot += red[w];
  const float inv = 1.0f / fmaxf(sqrtf(tot), 1e-8f);

  fn[(size_t)b * DFEAT + k0] = f0 * inv;
  fn[(size_t)b * DFEAT + k1] = f1 * inv;
}

// ---------------- cost = gamma * rownorm(1 - fnfn) + I ----------------
__global__ __launch_bounds__(128) void cost_kernel(const float* __restrict__ fnfn,
                                                   float* __restrict__ cost) {
  __shared__ float rmin[4], rmax[4];
  const int i = blockIdx.x;              // row
  const int j = threadIdx.x;             // col, 0..127
  const int lane = j & 31, wave = j >> 5;

  float raw = 1.0f - fnfn[(size_t)i * BATCH + j];
  float mn = wave_min32(raw);
  float mx = wave_max32(raw);
  if (lane == 0) { rmin[wave] = mn; rmax[wave] = mx; }
  __syncthreads();
  float cmin = fminf(fminf(rmin[0], rmin[1]), fminf(rmin[2], rmin[3]));
  float cmax = fmaxf(fmaxf(rmax[0], rmax[1]), fmaxf(rmax[2], rmax[3]));
  float c = 0.2f * (raw - cmin) / (cmax - cmin) + ((i == j) ? 1.0f : 0.0f);
  cost[(size_t)i * BATCH + j] = c;
}

// ---------------- fused CE + Sinkhorn + WCP : one block per problem m ----------------
__global__ __launch_bounds__(256) void sinkhorn_kernel(const float* __restrict__ gram,
                                                       const float* __restrict__ cost,
                                                       float* __restrict__ out) {
  extern __shared__ float smem[];
  float* cost_s   = smem;                     // [128*129]
  float* logits_s = cost_s + BATCH * CPAD;    // [128]
  float* logp1_s  = logits_s + BATCH;         // [128]
  float* u_s      = logp1_s + BATCH;          // [128]
  float* v_s      = u_s + BATCH;              // [128]
  float* red_s    = v_s + BATCH;              // [8]
  float* lse_sh   = red_s + 8;                // [1]

  const int m    = blockIdx.x;
  const int tid  = threadIdx.x;
  const int lane = tid & 31, wave = tid >> 5;
  const int i_row = m >> 2;
  const int col0  = (m & 3) * BATCH;

  // stage cost into LDS (padded rows)
  for (int e = tid; e < BATCH * BATCH; e += 256) {
    const int i = e >> 7, j = e & 127;
    cost_s[i * CPAD + j] = cost[e];
  }
  // logits[j] = -eudis(i_row, col0+j) = 2*G[i,j'] - G[i,i] - G[j',j']
  if (tid < BATCH) {
    const float g   = gram[(size_t)i_row * NTOT + col0 + tid];
    const float sqi = gram[(size_t)i_row * (NTOT + 1)];
    const float sqj = gram[(size_t)(col0 + tid) * (NTOT + 1)];
    logits_s[tid] = 2.0f * g - sqi - sqj;
    u_s[tid] = 0.0f;
    v_s[tid] = 0.0f;
  }
  __syncthreads();

  const float invT = 1.0f / 5.0f;                  // CE temperature
  const float invS = 0.044194173824159216f;        // 1/sqrt(512)

  if (wave == 0) {
    // CE at temp 5 (logsumexp over 128 cols)
    float mx = -INFINITY;
    #pragma unroll
    for (int q = 0; q < 4; ++q) mx = fmaxf(mx, logits_s[lane + 32 * q] * invT);
    mx = wave_max32(mx);
    float s = 0.0f;
    #pragma unroll
    for (int q = 0; q < 4; ++q) s += expf(logits_s[lane + 32 * q] * invT - mx);
    s = wave_sum32(s);
    const float lse5 = mx + logf(s);
    if (lane == 0) {
      const int tgt = i_row & (BATCH - 1);
      atomicAdd(out, (lse5 - logits_s[tgt] * invT) * (1.0f / MPROB));
    }
    // logsumexp at temp sqrt(d) for probs
    float mx2 = -INFINITY;
    #pragma unroll
    for (int q = 0; q < 4; ++q) mx2 = fmaxf(mx2, logits_s[lane + 32 * q] * invS);
    mx2 = wave_max32(mx2);
    float s2 = 0.0f;
    #pragma unroll
    for (int q = 0; q < 4; ++q) s2 += expf(logits_s[lane + 32 * q] * invS - mx2);
    s2 = wave_sum32(s2);
    if (lane == 0) lse_sh[0] = mx2 + logf(s2);
  }
  __syncthreads();
  if (tid < BATCH) {
    const float p = expf(logits_s[tid] * invS - lse_sh[0]);
    logp1_s[tid] = logf(p + 1e-12f);               // log(softmax + 1e-12)
  }
  __syncthreads();

  const float LOGP2 = -4.8520302639196171f;        // log(1/128 + 1e-12)

  #pragma unroll 1
  for (int it = 0; it < 5; ++it) {
    // u update: rows (lanes sweep j -> stride-1 LDS, conflict-free)
    for (int r = 0; r < 16; ++r) {
      const int i = wave * 16 + r;
      const float ui = u_s[i];
      float t[4], mx = -INFINITY;
      #pragma unroll
      for (int q = 0; q < 4; ++q) {
        const int j = lane + 32 * q;
        t[q] = (ui + v_s[j] - cost_s[i * CPAD + j]) * 2.0f;   // /eps, eps=0.5
        mx = fmaxf(mx, t[q]);
      }
      mx = wave_max32(mx);
      float s = 0.0f;
      #pragma unroll
      for (int q = 0; q < 4; ++q) s += expf(t[q] - mx);
      s = wave_sum32(s);
      if (lane == 0) u_s[i] = 0.5f * (logp1_s[i] - (mx + logf(s))) + ui;
    }
    __syncthreads();
    // v update: cols (lanes sweep i -> stride-129 LDS, conflict-free mod 64)
    for (int r = 0; r < 16; ++r) {
      const int j = wave * 16 + r;
      const float vj = v_s[j];
      float t[4], mx = -INFINITY;
      #pragma unroll
      for (int q = 0; q < 4; ++q) {
        const int i = lane + 32 * q;
        t[q] = (u_s[i] + vj - cost_s[i * CPAD + j]) * 2.0f;
        mx = fmaxf(mx, t[q]);
      }
      mx = wave_max32(mx);
      float s = 0.0f;
      #pragma unroll
      for (int q = 0; q < 4; ++q) s += expf(t[q] - mx);
      s = wave_sum32(s);
      if (lane == 0) v_s[j] = 0.5f * (LOGP2 - (mx + logf(s))) + vj;
    }
    __syncthreads();
  }

  // wcp contribution: sum_ij exp(M) * cost
  float acc = 0.0f;
  for (int e = tid; e < BATCH * BATCH; e += 256) {
    const int i = e >> 7, j = e & 127;
    const float cst = cost_s[i * CPAD + j];
    acc += expf((u_s[i] + v_s[j] - cst) * 2.0f) * cst;
  }
  acc = wave_sum32(acc);
  if (lane == 0) red_s[wave] = acc;
  __syncthreads();
  if (tid == 0) {
    float tot = 0.0f;
    #pragma unroll
    for (int w = 0; w < 8; ++w) tot += red_s[w];
    atomicAdd(out, tot * (1.0f / MPROB));
  }
}

// ---------------- launch ----------------
extern "C" void kernel_launch(void* const* d_in, const int* in_sizes, int n_in,
                              void* d_out, int out_size, void* d_ws, size_t ws_size,
                              hipStream_t stream) {
  (void)in_sizes; (void)n_in; (void)out_size; (void)ws_size;
  const float* feat = (const float*)d_in[0];
  float* out = (float*)d_out;

  // workspace layout (floats)
  float* ws    = (float*)d_ws;
  float* gram  = ws;                                  // 512*512
  float* fn    = gram + (size_t)NTOT * NTOT;          // 128*512
  float* fnfn  = fn + (size_t)BATCH * DFEAT;          // 128*128
  float* cost  = fnfn + (size_t)BATCH * BATCH;        // 128*128

  zero_out_kernel<<<1, 1, 0, stream>>>(out);

  // G = F F^T  (512x512, k=512) via fp32 WMMA
  gram_wmma_kernel<<<dim3(NTOT / 16, NTOT / 16), 32, 0, stream>>>(feat, gram, NTOT, DFEAT);

  // fn (mean over aug + L2 normalize)
  fn_kernel<<<BATCH, 256, 0, stream>>>(feat, fn);

  // fnfn = fn fn^T (128x128, k=512) via fp32 WMMA
  gram_wmma_kernel<<<dim3(BATCH / 16, BATCH / 16), 32, 0, stream>>>(fn, fnfn, BATCH, DFEAT);

  // cost matrix
  cost_kernel<<<BATCH, BATCH, 0, stream>>>(fnfn, cost);

  // fused CE + Sinkhorn + WCP, one block per problem
  const size_t shmem = (size_t)(BATCH * CPAD + 4 * BATCH + 8 + 1) * sizeof(float);
  sinkhorn_kernel<<<MPROB, 256, shmem, stream>>>(gram, cost, out);
}